// ThreeLevelCombiner_83769042141779
// MI455X (gfx1250) — compile-verified
//
#include <hip/hip_runtime.h>
#include <hip/hip_bf16.h>

typedef __attribute__((ext_vector_type(2))) float v2f;
typedef __attribute__((ext_vector_type(8))) float v8f;

#define T_L1 50
#define T_L2 60
#define T_SOMA 75
// DT = 1; tau_mem = 20 everywhere -> a = 0.05
// tau_syn: L1=5, L2=5, soma=8 ; tau_soft = 1 everywhere

__global__ __launch_bounds__(256) void ThreeLevelCombiner_83769042141779_kernel(
    const float* __restrict__ matcher,   // (B,16)
    const float* __restrict__ W_L1,      // (8,2)
    const float* __restrict__ W_L2,      // (16,8)
    const float* __restrict__ W_soma,    // (16,)
    float* __restrict__ out,             // soma(B) | L1(B,8) | L2(B,16)
    int B)
{
    __shared__ float s1[8][16][8];   // [wave][b_loc][k]   L1 spike times
    __shared__ float s2[8][16][16];  // [wave][b_loc][j]   L2 spike times

    const int lane = threadIdx.x & 31;
    const int wv   = threadIdx.x >> 5;
    const int batch_base = (blockIdx.x * 8 + wv) * 16;
    if (batch_base >= B) return;

    // ================= Level 1 =================
    // lane handles k = lane&7, batches b_loc = (lane>>3) + 4*i, i=0..3
    {
        const int k  = lane & 7;
        const float w0 = W_L1[2 * k];
        const float w1 = W_L1[2 * k + 1];
        float s0[4], s1v[4], E0[4], E1[4];
        #pragma unroll
        for (int i = 0; i < 4; ++i) {
            const int b = batch_base + (lane >> 3) + 4 * i;
            const float ta = matcher[b * 16 + 2 * k];
            const float tb = matcher[b * 16 + 2 * k + 1];
            s0[i] = ta; s1v[i] = tb;
            E0[i] = w0 * __expf(ta * 0.2f);   // exp(s/5)*w
            E1[i] = w1 * __expf(tb * 0.2f);
        }
        float v[4] = {0.f,0.f,0.f,0.f}, num[4] = {0.f,0.f,0.f,0.f}, den[4] = {0.f,0.f,0.f,0.f};
        for (int t = 0; t < T_L1; ++t) {
            const float tf = (float)t;
            const float et = __expf(-tf * 0.2f);  // exp(-t/5)
            #pragma unroll
            for (int i = 0; i < 4; ++i) {
                float p = (tf >= s0[i] ? E0[i] : 0.0f) + (tf >= s1v[i] ? E1[i] : 0.0f);
                float I = p * et;
                v[i] += (I - v[i]) * 0.05f;
                float e = __expf(v[i]);          // tau_soft = 1
                den[i] += e;
                num[i] = fmaf(tf, e, num[i]);
            }
        }
        #pragma unroll
        for (int i = 0; i < 4; ++i) {
            const int b_loc = (lane >> 3) + 4 * i;
            const float st = num[i] / den[i];
            s1[wv][b_loc][k] = st;
            out[B + (batch_base + b_loc) * 8 + k] = st;  // L1 output region
        }
    }
    __syncthreads();

    // ================= Level 2 (WMMA f32 16x16x4, chained K=0..3 / K=4..7) =====
    // A layout (ISA 16x4 f32): VGPR i, lane l -> M = l&15, K = i + 2*(l>>4)
    // C layout: VGPR r, lane l -> M = r + 8*(l>>4), N = l&15
    const int bm = lane & 15;     // M (batch row) for A, N (neuron col) for C
    const int hi = lane >> 4;
    {
        float sa[4], Ea[4];
        #pragma unroll
        for (int m = 0; m < 2; ++m)
            #pragma unroll
            for (int i = 0; i < 2; ++i) {
                const int kk = i + 2 * hi + 4 * m;
                const float st = s1[wv][bm][kk];
                sa[2 * m + i] = st;
                Ea[2 * m + i] = __expf(st * 0.2f);  // exp(s/5)
            }
        // B = W_L2^T (4x16 per WMMA): VGPR j, lane l -> K = j + 2*(l>>4), N = l&15
        v2f Bm1, Bm2;
        Bm1.x = W_L2[bm * 8 + (0 + 2 * hi)];
        Bm1.y = W_L2[bm * 8 + (1 + 2 * hi)];
        Bm2.x = W_L2[bm * 8 + (4 + 2 * hi)];
        Bm2.y = W_L2[bm * 8 + (5 + 2 * hi)];

        v8f vmem = {0.f,0.f,0.f,0.f,0.f,0.f,0.f,0.f};
        v8f num2 = {0.f,0.f,0.f,0.f,0.f,0.f,0.f,0.f};
        v8f den2 = {0.f,0.f,0.f,0.f,0.f,0.f,0.f,0.f};

        for (int t = 0; t < T_L2; ++t) {
            const float tf = (float)t;
            const float et = __expf(-tf * 0.2f);
            v2f A1, A2;
            A1.x = (tf >= sa[0]) ? Ea[0] * et : 0.0f;
            A1.y = (tf >= sa[1]) ? Ea[1] * et : 0.0f;
            A2.x = (tf >= sa[2]) ? Ea[2] * et : 0.0f;
            A2.y = (tf >= sa[3]) ? Ea[3] * et : 0.0f;
            v8f c = {0.f,0.f,0.f,0.f,0.f,0.f,0.f,0.f};
            c = __builtin_amdgcn_wmma_f32_16x16x4_f32(false, A1, false, Bm1,
                                                      (short)0, c, false, false);
            c = __builtin_amdgcn_wmma_f32_16x16x4_f32(false, A2, false, Bm2,
                                                      (short)0, c, false, false);
            #pragma unroll
            for (int r = 0; r < 8; ++r) {
                vmem[r] += (c[r] - vmem[r]) * 0.05f;
                float e = __expf(vmem[r]);
                den2[r] += e;
                num2[r] = fmaf(tf, e, num2[r]);
            }
        }
        #pragma unroll
        for (int r = 0; r < 8; ++r) {
            const int b_loc = r + 8 * hi;
            const float st = num2[r] / den2[r];
            s2[wv][b_loc][bm] = st;
            out[9 * B + (batch_base + b_loc) * 16 + bm] = st;  // L2 output region
        }
    }
    __syncthreads();

    // ================= Soma =================
    // lane pair (lane, lane^16) shares batch (lane&15); each half handles 8 synapses
    {
        const int sb = bm;            // batch index in tile
        float s3[8], E3[8];
        #pragma unroll
        for (int i = 0; i < 8; ++i) {
            const int j = hi * 8 + i;
            const float st = s2[wv][sb][j];
            s3[i] = st;
            E3[i] = W_soma[j] * __expf(st * 0.125f);   // exp(s/8) * w
        }
        float v3 = 0.f, num3 = 0.f, den3 = 0.f;
        for (int t = 0; t < T_SOMA; ++t) {
            const float tf = (float)t;
            const float et = __expf(-tf * 0.125f);     // exp(-t/8)
            float p = 0.f;
            #pragma unroll
            for (int i = 0; i < 8; ++i)
                p += (tf >= s3[i]) ? E3[i] : 0.0f;
            p *= et;
            p += __shfl_xor(p, 16, 32);                // combine the two synapse halves
            v3 += (p - v3) * 0.05f;
            float e = __expf(v3);
            den3 += e;
            num3 = fmaf(tf, e, num3);
        }
        if (hi == 0)
            out[batch_base + sb] = num3 / den3;        // soma output region
    }
}

extern "C" void kernel_launch(void* const* d_in, const int* in_sizes, int n_in,
                              void* d_out, int out_size, void* d_ws, size_t ws_size,
                              hipStream_t stream) {
    const float* matcher = (const float*)d_in[0];  // (B,16)
    const float* W_L1    = (const float*)d_in[1];  // (8,2)
    const float* W_L2    = (const float*)d_in[2];  // (16,8)
    const float* W_soma  = (const float*)d_in[3];  // (16,)
    float* out = (float*)d_out;
    const int B = in_sizes[0] / 16;

    const int batches_per_block = 128;  // 8 waves x 16 batches
    const int grid = (B + batches_per_block - 1) / batches_per_block;
    ThreeLevelCombiner_83769042141779_kernel<<<grid, 256, 0, stream>>>(
        matcher, W_L1, W_L2, W_soma, out, B);
}